// InvariantPointAttention_36240934044331
// MI455X (gfx1250) — compile-verified
//
#include <hip/hip_runtime.h>
#include <math.h>

#define NRES 1024
#define DCS  384
#define DCZ  128
#define DC   16
#define NH   12
#define NPQK 4
#define NPV  8
#define CAT  2112  // H*(CZ + C + PV*4)

typedef float v2f __attribute__((ext_vector_type(2)));
typedef float v8f __attribute__((ext_vector_type(8)));

static __device__ __forceinline__ v8f wmma4(v2f a, v2f b, v8f c) {
  // D = A(16x4 f32) * B(4x16 f32) + C(16x16 f32)
  return __builtin_amdgcn_wmma_f32_16x16x4_f32(false, a, false, b, (short)0, c, false, false);
}

// Async copy: 32 lanes x 16B -> 512B per wave-issue, LDS dest, tracked by ASYNCcnt.
static __device__ __forceinline__ void async_copy_b128(unsigned lds_byte_off,
                                                       const float* gaddr) {
  asm volatile("global_load_async_to_lds_b128 %0, %1, off"
               :: "v"(lds_byte_off), "v"(gaddr) : "memory");
}
#define ASYNC_WAIT(n) asm volatile("s_wait_asynccnt " #n ::: "memory")

// ---------------------------------------------------------------------------
// K1: linear projections q/k/v and rotated points (VALU; tiny fraction of time)
// layouts: q,k,v [n][h][c]; qp,kp [n][h][p*3+d] (12 floats); vp [n][h][p*3+d] (24)
// ---------------------------------------------------------------------------
__global__ __launch_bounds__(384)
void k_proj(const float* __restrict__ s, const float* __restrict__ rots,
            const float* __restrict__ trans,
            const float* __restrict__ w_q,  const float* __restrict__ b_q,
            const float* __restrict__ w_kv, const float* __restrict__ b_kv,
            const float* __restrict__ w_qp, const float* __restrict__ b_qp,
            const float* __restrict__ w_kvp,const float* __restrict__ b_kvp,
            float* __restrict__ q, float* __restrict__ k, float* __restrict__ v,
            float* __restrict__ qp, float* __restrict__ kp, float* __restrict__ vp) {
  __shared__ float s_s[DCS];
  __shared__ float s_raw[144 + 432];
  const int n = blockIdx.x, t = threadIdx.x;
  s_s[t] = s[(size_t)n * DCS + t];
  __syncthreads();

  for (int o = t; o < NH * DC; o += blockDim.x) {
    const float* w = w_q + (size_t)o * DCS;
    float acc = b_q[o];
    #pragma unroll 8
    for (int i = 0; i < DCS; ++i) acc = fmaf(s_s[i], w[i], acc);
    q[(size_t)n * NH * DC + o] = acc;
  }
  for (int o = t; o < 2 * NH * DC; o += blockDim.x) {
    const float* w = w_kv + (size_t)o * DCS;
    float acc = b_kv[o];
    #pragma unroll 8
    for (int i = 0; i < DCS; ++i) acc = fmaf(s_s[i], w[i], acc);
    const int h = o >> 5, c2 = o & 31;
    if (c2 < DC) k[((size_t)n * NH + h) * DC + c2]        = acc;
    else         v[((size_t)n * NH + h) * DC + (c2 - DC)] = acc;
  }
  for (int o = t; o < 144; o += blockDim.x) {
    const float* w = w_qp + (size_t)o * DCS;
    float acc = b_qp[o];
    #pragma unroll 8
    for (int i = 0; i < DCS; ++i) acc = fmaf(s_s[i], w[i], acc);
    s_raw[o] = acc;
  }
  for (int o = t; o < 432; o += blockDim.x) {
    const float* w = w_kvp + (size_t)o * DCS;
    float acc = b_kvp[o];
    #pragma unroll 8
    for (int i = 0; i < DCS; ++i) acc = fmaf(s_s[i], w[i], acc);
    s_raw[144 + o] = acc;
  }
  __syncthreads();

  const float* R  = rots + (size_t)n * 9;
  const float* tr = trans + (size_t)n * 3;
  for (int pidx = t; pidx < 48; pidx += blockDim.x) {
    const float px = s_raw[0 * 48 + pidx], py = s_raw[1 * 48 + pidx], pz = s_raw[2 * 48 + pidx];
    const int h = pidx / NPQK, p = pidx % NPQK;
    float* dst = qp + ((size_t)n * NH + h) * (NPQK * 3) + p * 3;
    dst[0] = R[0] * px + R[1] * py + R[2] * pz + tr[0];
    dst[1] = R[3] * px + R[4] * py + R[5] * pz + tr[1];
    dst[2] = R[6] * px + R[7] * py + R[8] * pz + tr[2];
  }
  for (int pidx = t; pidx < 144; pidx += blockDim.x) {
    const float px = s_raw[144 + 0 * 144 + pidx];
    const float py = s_raw[144 + 1 * 144 + pidx];
    const float pz = s_raw[144 + 2 * 144 + pidx];
    const int h = pidx / 12, p = pidx % 12;
    float ox = R[0] * px + R[1] * py + R[2] * pz + tr[0];
    float oy = R[3] * px + R[4] * py + R[5] * pz + tr[1];
    float oz = R[6] * px + R[7] * py + R[8] * pz + tr[2];
    float* dst = (p < NPQK)
        ? (kp + ((size_t)n * NH + h) * (NPQK * 3) + p * 3)
        : (vp + ((size_t)n * NH + h) * (NPV * 3) + (p - NPQK) * 3);
    dst[0] = ox; dst[1] = oy; dst[2] = oz;
  }
}

// ---------------------------------------------------------------------------
// K2: fused attention logits. One wave per 16x16 (i,j) tile, all 12 heads.
// First (and only) z read of this phase goes straight into WMMA A operands.
// attn[h][i][j] = qk/sqrt(48) + (bpair+b_b)/sqrt(3) - 0.5*hw*d2 + mask_term
// ---------------------------------------------------------------------------
__global__ __launch_bounds__(32)
void k_logits(const float* __restrict__ q,  const float* __restrict__ k,
              const float* __restrict__ qp, const float* __restrict__ kp,
              const float* __restrict__ z,  const float* __restrict__ w_b,
              const float* __restrict__ b_b,
              const float* __restrict__ hwts, const float* __restrict__ mask,
              float* __restrict__ attn) {
  __shared__ float s_bp[16][16][NH];  // [i_local][j_local][h]
  __shared__ float s_wbT[DCZ][16];    // w_b transposed, padded to 16 heads
  __shared__ float s_qn[16], s_kn[16];

  const int jt = blockIdx.x, it = blockIdx.y;
  const int i0 = it * 16, j0 = jt * 16;
  const int lane = threadIdx.x;
  const int half = lane >> 4;
  const int l16  = lane & 15;

  for (int idx = lane; idx < DCZ * 16; idx += 32) {
    const int c = idx >> 4, h = idx & 15;
    s_wbT[c][h] = (h < NH) ? w_b[h * DCZ + c] : 0.f;
  }
  __syncthreads();

  // ---- bpair: for each i in tile, GEMM (M=16 j's) x (N=heads) over K=128 ----
  for (int g = 0; g < 16; ++g) {
    const float* zr = z + ((size_t)(i0 + g) * NRES + (size_t)(j0 + l16)) * DCZ;
    if (g + 2 < 16) {
      // pull the A-operand rows for iteration g+2 toward L2 (global_prefetch_b8)
      __builtin_prefetch(z + ((size_t)(i0 + g + 2) * NRES + (size_t)(j0 + l16)) * DCZ, 0, 1);
    }
    v8f acc = {};
    #pragma unroll
    for (int st = 0; st < 32; ++st) {
      const int kb = st * 4 + 2 * half;
      v2f a = *(const v2f*)(zr + kb);                          // A[m=j_local][k]
      v2f b; b.x = s_wbT[kb][l16]; b.y = s_wbT[kb + 1][l16];   // B[k][n=h]
      acc = wmma4(a, b, acc);
    }
    #pragma unroll
    for (int r = 0; r < 8; ++r) {
      const int jl = r + 8 * half;
      if (l16 < NH) s_bp[g][jl][l16] = acc[r];
    }
  }
  __syncthreads();

  const float scale_qk = 0.14433756729740646f; // sqrt(1/48)
  const float scale_b  = 0.5773502691896258f;  // sqrt(1/3)
  const float scale_pt = 0.13608276348795434f; // sqrt(1/54)

  for (int h = 0; h < NH; ++h) {
    {
      const float* pr = (half == 0)
          ? (qp + ((size_t)(i0 + l16) * NH + h) * 12)
          : (kp + ((size_t)(j0 + l16) * NH + h) * 12);
      float pn = 0.f;
      #pragma unroll
      for (int d = 0; d < 12; ++d) pn = fmaf(pr[d], pr[d], pn);
      if (half == 0) s_qn[l16] = pn; else s_kn[l16] = pn;
    }
    __syncthreads();

    v8f cqk = {};
    #pragma unroll
    for (int st = 0; st < 4; ++st) {
      const int kb = st * 4 + 2 * half;
      v2f a = *(const v2f*)(q + ((size_t)(i0 + l16) * NH + h) * DC + kb);
      v2f b = *(const v2f*)(k + ((size_t)(j0 + l16) * NH + h) * DC + kb);
      cqk = wmma4(a, b, cqk);
    }
    v8f cpt = {};
    #pragma unroll
    for (int st = 0; st < 3; ++st) {
      const int kb = st * 4 + 2 * half;
      v2f a = *(const v2f*)(qp + ((size_t)(i0 + l16) * NH + h) * 12 + kb);
      v2f b = *(const v2f*)(kp + ((size_t)(j0 + l16) * NH + h) * 12 + kb);
      cpt = wmma4(a, b, cpt);
    }

    const float hw = logf(1.f + __expf(hwts[h])) * scale_pt;  // softplus * sqrt(1/54)
    const float bb = b_b[h];
    const float mj = mask[j0 + l16];
    #pragma unroll
    for (int r = 0; r < 8; ++r) {
      const int mi = r + 8 * half;
      const float d2 = s_qn[mi] + s_kn[l16] - 2.f * cpt[r];
      const float lg = cqk[r] * scale_qk + (s_bp[mi][l16][h] + bb) * scale_b
                     - 0.5f * hw * d2
                     + (mask[i0 + mi] * mj - 1.f) * 100000.f;
      attn[((size_t)h * NRES + (i0 + mi)) * NRES + (j0 + l16)] = lg;
    }
    __syncthreads();
  }
}

// ---------------------------------------------------------------------------
// K3: row softmax over j (rows attn[h][i][:])
// ---------------------------------------------------------------------------
__global__ __launch_bounds__(256)
void k_softmax(float* __restrict__ attn) {
  __shared__ float red[256];
  float* a = attn + (size_t)blockIdx.x * NRES;
  const int t = threadIdx.x;
  float x0 = a[t], x1 = a[t + 256], x2 = a[t + 512], x3 = a[t + 768];
  float m = fmaxf(fmaxf(x0, x1), fmaxf(x2, x3));
  red[t] = m; __syncthreads();
  for (int s2 = 128; s2 > 0; s2 >>= 1) { if (t < s2) red[t] = fmaxf(red[t], red[t + s2]); __syncthreads(); }
  m = red[0]; __syncthreads();
  x0 = __expf(x0 - m); x1 = __expf(x1 - m); x2 = __expf(x2 - m); x3 = __expf(x3 - m);
  red[t] = x0 + x1 + x2 + x3; __syncthreads();
  for (int s2 = 128; s2 > 0; s2 >>= 1) { if (t < s2) red[t] += red[t + s2]; __syncthreads(); }
  const float inv = 1.f / red[0];
  a[t] = x0 * inv; a[t + 256] = x1 * inv; a[t + 512] = x2 * inv; a[t + 768] = x3 * inv;
}

// ---------------------------------------------------------------------------
// K4: o = A@V and o_pt = A@v_pts via WMMA. One wave per (head, i-tile).
// A operand (attention row tile) loaded once and reused across 3 N-tiles.
// ---------------------------------------------------------------------------
__global__ __launch_bounds__(32)
void k_av(const float* __restrict__ attn, const float* __restrict__ v,
          const float* __restrict__ vp,
          float* __restrict__ oacc, float* __restrict__ optacc) {
  const int h  = blockIdx.x;
  const int i0 = blockIdx.y * 16;
  const int lane = threadIdx.x, half = lane >> 4, l16 = lane & 15;
  const int n1 = 16 + l16;
  v8f co = {}, cp0 = {}, cp1 = {};
  for (int j0 = 0; j0 < NRES; j0 += 16) {
    #pragma unroll
    for (int st = 0; st < 4; ++st) {
      const int kb = st * 4 + 2 * half;
      v2f a = *(const v2f*)(attn + ((size_t)h * NRES + (i0 + l16)) * NRES + j0 + kb);
      v2f b;
      b.x = v[((size_t)(j0 + kb)     * NH + h) * DC + l16];
      b.y = v[((size_t)(j0 + kb + 1) * NH + h) * DC + l16];
      co = wmma4(a, b, co);
      const float* vr0 = vp + ((size_t)(j0 + kb)     * NH + h) * 24;
      const float* vr1 = vp + ((size_t)(j0 + kb + 1) * NH + h) * 24;
      v2f b0; b0.x = vr0[l16]; b0.y = vr1[l16];
      cp0 = wmma4(a, b0, cp0);
      v2f b1; b1.x = (n1 < 24) ? vr0[n1] : 0.f; b1.y = (n1 < 24) ? vr1[n1] : 0.f;
      cp1 = wmma4(a, b1, cp1);
    }
  }
  #pragma unroll
  for (int r = 0; r < 8; ++r) {
    const int mi = r + 8 * half;
    oacc[((size_t)(i0 + mi) * NH + h) * DC + l16] = co[r];
    optacc[((size_t)(i0 + mi) * NH + h) * 24 + l16] = cp0[r];
    if (n1 < 24) optacc[((size_t)(i0 + mi) * NH + h) * 24 + n1] = cp1[r];
  }
}

// ---------------------------------------------------------------------------
// K4p: o_pair[i,h,c] = sum_j a[h,i,j] * z[i,j,c].  B-operand varies with the
// output row -> not a WMMA GEMM. Second (final) z pass:
//   * z rows streamed into LDS with GLOBAL_LOAD_ASYNC_TO_LDS_B128,
//     double-buffered, drained with s_wait_asynccnt (ASYNCcnt path).
//   * attention probs are block-uniform -> read directly so the compiler
//     scalarizes them (s_load + v_fmac v,s,v), no LDS broadcast traffic.
// Per (j,c): 1 LDS read + 12 scalar-operand FMAs.
// ---------------------------------------------------------------------------
__global__ __launch_bounds__(128)
void k_opair(const float* __restrict__ attn, const float* __restrict__ z,
             float* __restrict__ opair) {
  __shared__ float zt[2][32][DCZ];               // 2 x 16 KB chunk buffers
  const int i = blockIdx.x;
  const int t = threadIdx.x;                     // 128 threads = 4 waves
  const int w = t >> 5, l = t & 31;

  float acc[NH];
  #pragma unroll
  for (int h = 0; h < NH; ++h) acc[h] = 0.f;

  const float* zrow0 = z + (size_t)i * NRES * DCZ;

  // stage rows [jb, jb+32) into buffer `buf`: wave w issues rows w*8 .. w*8+7,
  // each async issue moves one 512B z row (32 lanes x 16B).
  auto stage = [&](int jb, int buf) {
    #pragma unroll
    for (int s2 = 0; s2 < 8; ++s2) {
      const int jj = w * 8 + s2;
      const unsigned lds_off =
          (unsigned)(uintptr_t)(&zt[buf][jj][0]) + (unsigned)(l * 16);
      const float* g = zrow0 + (size_t)(jb + jj) * DCZ + l * 4;
      async_copy_b128(lds_off, g);
    }
  };

  stage(0, 0);
  int buf = 0;
  for (int jb = 0; jb < NRES; jb += 32) {
    if (jb + 32 < NRES) {
      stage(jb + 32, buf ^ 1);
      ASYNC_WAIT(8);    // drain current chunk; next chunk (8 issues) in flight
    } else {
      ASYNC_WAIT(0);
    }
    __syncthreads();

    const float* ap = attn + (size_t)i * NRES + jb;   // block-uniform pointer
    #pragma unroll 4
    for (int jj = 0; jj < 32; ++jj) {
      const float zv = zt[buf][jj][t];
      #pragma unroll
      for (int h = 0; h < NH; ++h)
        acc[h] = fmaf(ap[(size_t)h * NRES * NRES + jj], zv, acc[h]);
    }
    __syncthreads();
    buf ^= 1;
  }
  #pragma unroll
  for (int h = 0; h < NH; ++h)
    opair[((size_t)i * NH + h) * DCZ + t] = acc[h];
}

// ---------------------------------------------------------------------------
// K4b: inverse-frame rotation of o_pt, norms, and assembly of o_cat (2112/row)
// o_cat = [o(192) | pt.x(96) | pt.y(96) | pt.z(96) | |pt|(96) | o_pair(1536)]
// ---------------------------------------------------------------------------
__global__ __launch_bounds__(128)
void k_assemble(const float* __restrict__ oacc, const float* __restrict__ optacc,
                const float* __restrict__ opair, const float* __restrict__ rots,
                const float* __restrict__ trans, float* __restrict__ ocat) {
  const int n = blockIdx.x, t = threadIdx.x;
  float* dst = ocat + (size_t)n * CAT;
  for (int o = t; o < NH * DC; o += blockDim.x) dst[o] = oacc[(size_t)n * NH * DC + o];
  const float* R  = rots + (size_t)n * 9;
  const float* tr = trans + (size_t)n * 3;
  for (int pidx = t; pidx < NH * NPV; pidx += blockDim.x) {
    const int h = pidx / NPV, p = pidx % NPV;
    const float* pa = optacc + ((size_t)n * NH + h) * 24 + p * 3;
    const float x = pa[0] - tr[0], y = pa[1] - tr[1], zc = pa[2] - tr[2];
    const float ox = R[0] * x + R[3] * y + R[6] * zc;   // R^T * (acc - t)
    const float oy = R[1] * x + R[4] * y + R[7] * zc;
    const float oz = R[2] * x + R[5] * y + R[8] * zc;
    dst[192 + 0 * 96 + pidx] = ox;
    dst[192 + 1 * 96 + pidx] = oy;
    dst[192 + 2 * 96 + pidx] = oz;
    dst[192 + 288 + pidx]    = sqrtf(ox * ox + oy * oy + oz * oz + 1e-8f);
  }
  for (int o = t; o < NH * DCZ; o += blockDim.x)
    dst[576 + o] = opair[(size_t)n * NH * DCZ + o];
}

// ---------------------------------------------------------------------------
// K5: out = o_cat @ w_out.T + b_out  (1024 x 384, K=2112) via WMMA
// ---------------------------------------------------------------------------
__global__ __launch_bounds__(32)
void k_out(const float* __restrict__ ocat, const float* __restrict__ w_out,
           const float* __restrict__ b_out, float* __restrict__ out) {
  const int n0 = blockIdx.x * 16, i0 = blockIdx.y * 16;
  const int lane = threadIdx.x, half = lane >> 4, l16 = lane & 15;
  const float* ar = ocat  + (size_t)(i0 + l16) * CAT;
  const float* br = w_out + (size_t)(n0 + l16) * CAT;
  v8f c = {};
  #pragma unroll 4
  for (int kk = 0; kk < CAT; kk += 4) {
    const int kb = kk + 2 * half;
    v2f a = *(const v2f*)(ar + kb);
    v2f b = *(const v2f*)(br + kb);
    c = wmma4(a, b, c);
  }
  const float bo = b_out[n0 + l16];
  #pragma unroll
  for (int r = 0; r < 8; ++r) {
    const int mi = r + 8 * half;
    out[(size_t)(i0 + mi) * DCS + n0 + l16] = c[r] + bo;
  }
}

// ---------------------------------------------------------------------------
extern "C" void kernel_launch(void* const* d_in, const int* in_sizes, int n_in,
                              void* d_out, int out_size, void* d_ws, size_t ws_size,
                              hipStream_t stream) {
  const float* s     = (const float*)d_in[0];
  const float* z     = (const float*)d_in[1];
  const float* rots  = (const float*)d_in[2];
  const float* trans = (const float*)d_in[3];
  const float* mask  = (const float*)d_in[4];
  const float* w_q   = (const float*)d_in[5];
  const float* b_q   = (const float*)d_in[6];
  const float* w_b   = (const float*)d_in[7];
  const float* b_b   = (const float*)d_in[8];
  const float* w_kv  = (const float*)d_in[9];
  const float* b_kv  = (const float*)d_in[10];
  const float* w_qp  = (const float*)d_in[11];
  const float* b_qp  = (const float*)d_in[12];
  const float* w_kvp = (const float*)d_in[13];
  const float* b_kvp = (const float*)d_in[14];
  const float* hwts  = (const float*)d_in[15];
  const float* w_out = (const float*)d_in[16];
  const float* b_out = (const float*)d_in[17];

  // workspace carve-up (fp32). Total ~18.0M floats ~= 72 MB.
  float* ws = (float*)d_ws;
  float* q      = ws; ws += NRES * NH * DC;
  float* k      = ws; ws += NRES * NH * DC;
  float* v      = ws; ws += NRES * NH * DC;
  float* qp     = ws; ws += NRES * NH * NPQK * 3;
  float* kp     = ws; ws += NRES * NH * NPQK * 3;
  float* vp     = ws; ws += NRES * NH * NPV * 3;
  float* attn   = ws; ws += (size_t)NH * NRES * NRES;
  float* oacc   = ws; ws += NRES * NH * DC;
  float* optacc = ws; ws += NRES * NH * NPV * 3;
  float* opair  = ws; ws += NRES * NH * DCZ;
  float* ocat   = ws; ws += (size_t)NRES * CAT;

  k_proj<<<NRES, 384, 0, stream>>>(s, rots, trans, w_q, b_q, w_kv, b_kv,
                                   w_qp, b_qp, w_kvp, b_kvp,
                                   q, k, v, qp, kp, vp);
  k_logits<<<dim3(NRES / 16, NRES / 16), 32, 0, stream>>>(q, k, qp, kp, z, w_b, b_b,
                                                          hwts, mask, attn);
  k_softmax<<<NH * NRES, 256, 0, stream>>>(attn);
  k_av<<<dim3(NH, NRES / 16), 32, 0, stream>>>(attn, v, vp, oacc, optacc);
  k_opair<<<NRES, 128, 0, stream>>>(attn, z, opair);
  k_assemble<<<NRES, 128, 0, stream>>>(oacc, optacc, opair, rots, trans, ocat);
  k_out<<<dim3(DCS / 16, NRES / 16), 32, 0, stream>>>(ocat, w_out, b_out, (float*)d_out);
}